// AllegroLayer_56504589746529
// MI455X (gfx1250) — compile-verified
//
#include <hip/hip_runtime.h>

#define E_EDGES 16384
#define N_NODES 1024
#define NMUL 64
#define FIN 64

typedef float v2f __attribute__((ext_vector_type(2)));
typedef float v8f __attribute__((ext_vector_type(8)));

// ======================= compile-time Clebsch-Gordan =======================
constexpr double d_sqrt(double x) {
  double g = (x > 1.0) ? x : 1.0;
  for (int i = 0; i < 64; ++i) g = 0.5 * (g + x / g);
  return g;
}
constexpr double PI_D = 3.14159265358979323846264338327950288;
constexpr double S3v  = d_sqrt(3.0);
constexpr double S5v  = d_sqrt(5.0);
constexpr double S15v = d_sqrt(15.0);
constexpr double C33v = d_sqrt(70.0) / 4.0;
constexpr double C32v = d_sqrt(105.0) / 2.0;
constexpr double C31v = d_sqrt(42.0) / 4.0;
constexpr double C30v = d_sqrt(7.0) / 2.0;
constexpr double CXYZv = d_sqrt(105.0);

struct Term { double c; int a, b, d; };
struct Poly { int n; Term t[2]; };

constexpr Poly shp(int l, int m) {
  if (l == 0) return Poly{1, {Term{1.0, 0, 0, 0}, Term{0, 0, 0, 0}}};
  if (l == 1) {
    if (m == 0) return Poly{1, {Term{S3v, 1, 0, 0}, Term{0, 0, 0, 0}}};
    if (m == 1) return Poly{1, {Term{S3v, 0, 1, 0}, Term{0, 0, 0, 0}}};
    return Poly{1, {Term{S3v, 0, 0, 1}, Term{0, 0, 0, 0}}};
  }
  if (l == 2) {
    if (m == 0) return Poly{1, {Term{S15v, 1, 1, 0}, Term{0, 0, 0, 0}}};
    if (m == 1) return Poly{1, {Term{S15v, 0, 1, 1}, Term{0, 0, 0, 0}}};
    if (m == 2) return Poly{2, {Term{1.5 * S5v, 0, 0, 2}, Term{-0.5 * S5v, 0, 0, 0}}};
    if (m == 3) return Poly{1, {Term{S15v, 1, 0, 1}, Term{0, 0, 0, 0}}};
    return Poly{2, {Term{0.5 * S15v, 2, 0, 0}, Term{-0.5 * S15v, 0, 2, 0}}};
  }
  if (m == 0) return Poly{2, {Term{3.0 * C33v, 2, 1, 0}, Term{-C33v, 0, 3, 0}}};
  if (m == 1) return Poly{1, {Term{CXYZv, 1, 1, 1}, Term{0, 0, 0, 0}}};
  if (m == 2) return Poly{2, {Term{5.0 * C31v, 0, 1, 2}, Term{-C31v, 0, 1, 0}}};
  if (m == 3) return Poly{2, {Term{5.0 * C30v, 0, 0, 3}, Term{-3.0 * C30v, 0, 0, 1}}};
  if (m == 4) return Poly{2, {Term{5.0 * C31v, 1, 0, 2}, Term{-C31v, 1, 0, 0}}};
  if (m == 5) return Poly{2, {Term{C32v, 2, 0, 1}, Term{-C32v, 0, 2, 1}}};
  return Poly{2, {Term{C33v, 3, 0, 0}, Term{-3.0 * C33v, 1, 2, 0}}};
}

constexpr double dfact(int n) { double r = 1.0; while (n > 1) { r *= n; n -= 2; } return r; }

constexpr double mono_int(int a, int b, int c) {
  if ((a & 1) || (b & 1) || (c & 1)) return 0.0;
  return 4.0 * PI_D * dfact(a - 1) * dfact(b - 1) * dfact(c - 1) / dfact(a + b + c + 1);
}

constexpr double cg_raw(int l1, int i, int l2, int j, int l3, int k) {
  Poly p1 = shp(l1, i), p2 = shp(l2, j), p3 = shp(l3, k);
  double s = 0.0;
  for (int t1 = 0; t1 < p1.n; ++t1)
    for (int t2 = 0; t2 < p2.n; ++t2)
      for (int t3 = 0; t3 < p3.n; ++t3) {
        s += p1.t[t1].c * p2.t[t2].c * p3.t[t3].c *
             mono_int(p1.t[t1].a + p2.t[t2].a + p3.t[t3].a,
                      p1.t[t1].b + p2.t[t2].b + p3.t[t3].b,
                      p1.t[t1].d + p2.t[t2].d + p3.t[t3].d);
      }
  return s / (4.0 * PI_D);
}

template <int L1, int L2, int L3>
struct CGT { float v[2 * L1 + 1][2 * L2 + 1][2 * L3 + 1]; };

template <int L1, int L2, int L3>
constexpr CGT<L1, L2, L3> make_cg() {
  CGT<L1, L2, L3> r{};
  double nrm = 0.0;
  for (int i = 0; i < 2 * L1 + 1; ++i)
    for (int j = 0; j < 2 * L2 + 1; ++j)
      for (int k = 0; k < 2 * L3 + 1; ++k) {
        double c = cg_raw(L1, i, L2, j, L3, k);
        nrm += c * c;
      }
  double scale = d_sqrt((double)(2 * L3 + 1)) / d_sqrt(nrm);
  for (int i = 0; i < 2 * L1 + 1; ++i)
    for (int j = 0; j < 2 * L2 + 1; ++j)
      for (int k = 0; k < 2 * L3 + 1; ++k)
        r.v[i][j][k] = (float)(cg_raw(L1, i, L2, j, L3, k) * scale);
  return r;
}

// Fully unrolled contraction: out[k] = sum_ij CG[i][j][k]*A[i]*B[j] (zeros fold away)
template <int L1, int L2, int L3>
__device__ __forceinline__ void tp_acc(const float* A, const float* B, float* out) {
  constexpr CGT<L1, L2, L3> cg = make_cg<L1, L2, L3>();
#pragma unroll
  for (int k = 0; k < 2 * L3 + 1; ++k) {
    float s = 0.f;
#pragma unroll
    for (int i = 0; i < 2 * L1 + 1; ++i)
#pragma unroll
      for (int j = 0; j < 2 * L2 + 1; ++j)
        if (cg.v[i][j][k] != 0.f) s = fmaf(cg.v[i][j][k] * A[i], B[j], s);
    out[k] = s;
  }
}

// ============================ WMMA helpers ================================
__device__ __forceinline__ v8f vzero() {
  v8f z;
#pragma unroll
  for (int i = 0; i < 8; ++i) z[i] = 0.f;
  return z;
}

__device__ __forceinline__ v8f wmma4(v2f a, v2f b, v8f c) {
  // D = A(16x4,f32) * B(4x16,f32) + C(16x16,f32)
  return __builtin_amdgcn_wmma_f32_16x16x4_f32(false, a, false, b, (short)0, c, false, false);
}

__device__ __forceinline__ float nsilu(float v) {
  // silu(v) / ACT_C ; ACT_C = sqrt(E[silu(z)^2]) ~= 0.5967
  return (v / (1.f + __expf(-v))) * 1.675883f;
}

// ============================== kernels ===================================
__global__ void k_init(float* __restrict__ nodebuf) {
  int i = blockIdx.x * blockDim.x + threadIdx.x;
  if (i < N_NODES * NMUL * 16) nodebuf[i] = 0.f;
}

// Pre-pack a (K x 64) row-major weight matrix into WMMA B-fragment order:
// pOut[(k0/4)*2 + h][n] = float2( W[(k0+2h)*64+n], W[(k0+2h+1)*64+n] )
// -> each lane's B fragment becomes one global_load_b64.
__global__ void k_pack(const float* __restrict__ W, int K, float* __restrict__ pOut) {
  int idx = blockIdx.x * blockDim.x + threadIdx.x;  // (K/4)*2*64 fragments
  if (idx >= (K >> 2) * 128) return;
  int n  = idx & 63;
  int h  = (idx >> 6) & 1;
  int k0 = (idx >> 7) << 2;
  v2f b;
  b.x = W[(k0 + 2 * h) * 64 + n];
  b.y = W[(k0 + 2 * h + 1) * 64 + n];
  *(v2f*)(pOut + (size_t)idx * 2) = b;
}

// w = x @ W_w / 8   (M=E, N=64, K=64); one wave per 16-edge tile; packed B
__global__ void k_w(const float* __restrict__ x, const float* __restrict__ pWw,
                    float* __restrict__ w) {
  int e0 = blockIdx.x * 16;
  int lane = threadIdx.x & 31;
  int r = lane & 15, h = lane >> 4;
  const float* A = x + (size_t)e0 * FIN;
  const v2f* pB = (const v2f*)pWw;
  v8f acc[4] = {vzero(), vzero(), vzero(), vzero()};
  for (int k0 = 0; k0 < FIN; k0 += 4) {
    v2f a = *(const v2f*)(A + r * FIN + k0 + 2 * h);
    int bbase = (((k0 >> 2) * 2 + h) << 6) + r;
#pragma unroll
    for (int nt = 0; nt < 4; ++nt) {
      v2f b = pB[bbase + nt * 16];
      acc[nt] = wmma4(a, b, acc[nt]);
    }
  }
#pragma unroll
  for (int nt = 0; nt < 4; ++nt)
#pragma unroll
    for (int v = 0; v < 8; ++v)
      w[(size_t)(e0 + v + 8 * h) * NMUL + nt * 16 + r] = acc[nt][v] * 0.125f;
}

// Per edge: d, Y(16), scatter-add w*Y into node accumulator
__global__ void k_edge(const float* __restrict__ vec, const int* __restrict__ senders,
                       const float* __restrict__ w, float* __restrict__ nodebuf,
                       float* __restrict__ dws) {
  __shared__ float sY[16];
  int e = blockIdx.x;
  int t = threadIdx.x;  // 128 threads
  if (t == 0) {
    float vx = vec[e * 3 + 0], vy = vec[e * 3 + 1], vz = vec[e * 3 + 2];
    float d = sqrtf(vx * vx + vy * vy + vz * vz);
    dws[e] = d;
    float inv = 1.f / d;
    float X = vx * inv, Y = vy * inv, Z = vz * inv;
    const float fS3 = (float)S3v, fS5 = (float)S5v, fS15 = (float)S15v;
    const float f33 = (float)C33v, f32c = (float)C32v, f31 = (float)C31v;
    const float f30 = (float)C30v, fXYZ = (float)CXYZv;
    sY[0] = 1.f;
    sY[1] = fS3 * X;  sY[2] = fS3 * Y;  sY[3] = fS3 * Z;
    sY[4] = fS15 * X * Y;
    sY[5] = fS15 * Y * Z;
    sY[6] = 0.5f * fS5 * (3.f * Z * Z - 1.f);
    sY[7] = fS15 * X * Z;
    sY[8] = 0.5f * fS15 * (X * X - Y * Y);
    sY[9]  = f33 * Y * (3.f * X * X - Y * Y);
    sY[10] = fXYZ * X * Y * Z;
    sY[11] = f31 * Y * (5.f * Z * Z - 1.f);
    sY[12] = f30 * Z * (5.f * Z * Z - 3.f);
    sY[13] = f31 * X * (5.f * Z * Z - 1.f);
    sY[14] = f32c * Z * (X * X - Y * Y);
    sY[15] = f33 * X * (X * X - 3.f * Y * Y);
  }
  __syncthreads();
  int s = senders[e];
  float* nb = nodebuf + (size_t)s * NMUL * 16;
  const float* we = w + (size_t)e * NMUL;
#pragma unroll
  for (int i = 0; i < 8; ++i) {
    int p = t + i * 128;            // 1024 (n,k) pairs
    int n = p >> 4, k = p & 15;
    atomicAdd(&nb[n * 16 + k], we[n] * sY[k]);
  }
}

// Fused: gather agg -> CG tensor products (LDS) -> 8 output WMMA GEMMs + 3-stage WMMA MLP
__global__ void __launch_bounds__(256)
k_fused(const float* __restrict__ xg, const float* __restrict__ Vg,
        const int* __restrict__ senders, const float* __restrict__ nodebuf,
        const float* __restrict__ dws,
        const float* __restrict__ pW1, const float* __restrict__ pW2,
        const float* __restrict__ pW3,
        const float* __restrict__ pWv1, const float* __restrict__ pWv2,
        float* __restrict__ xout, float* __restrict__ Vout) {
  extern __shared__ float lds[];
  float* h0  = lds;            // 16 x 256
  float* lv1 = lds + 4096;     // 3 comps x 16 edges x 320
  float* lv2 = lv1 + 15360;    // 5 comps x 16 edges x 320
  float* hT  = lv2 + 25600;    // 16 x 64
  float* hT2 = hT + 1024;      // 16 x 64

  int e0 = blockIdx.x * 16;
  int tid = threadIdx.x;       // 256 threads = 8 waves

  // ---- phase A: tensor products into LDS (4 (edge,n) pairs per thread) ----
  for (int j = 0; j < 4; ++j) {
    int q = tid + j * 256;
    int et = q >> 6, n = q & 63;
    int e = e0 + et;
    int s = senders[e];
    float agg[16];
    const float* nb = nodebuf + ((size_t)s * NMUL + n) * 16;
#pragma unroll
    for (int k = 0; k < 16; ++k) agg[k] = nb[k] * 0.25f;  // 1/sqrt(AVG_NEIGH)
    float Vv[9];
    const float* vp = Vg + ((size_t)e * NMUL + n) * 9;
#pragma unroll
    for (int k = 0; k < 9; ++k) Vv[k] = vp[k];

    const float* A0 = agg;     const float* A1 = agg + 1;
    const float* A2 = agg + 4; const float* A3 = agg + 9;
    const float* B0 = Vv;      const float* B1 = Vv + 1;  const float* B2 = Vv + 4;

    float sc0[1], sc1[1], sc2[1];
    tp_acc<0, 0, 0>(A0, B0, sc0);
    tp_acc<1, 1, 0>(A1, B1, sc1);
    tp_acc<2, 2, 0>(A2, B2, sc2);
    float v1[5][3], v2[5][5];
    tp_acc<0, 1, 1>(A0, B1, v1[0]);
    tp_acc<1, 0, 1>(A1, B0, v1[1]);
    tp_acc<1, 2, 1>(A1, B2, v1[2]);
    tp_acc<2, 1, 1>(A2, B1, v1[3]);
    tp_acc<3, 2, 1>(A3, B2, v1[4]);
    tp_acc<0, 2, 2>(A0, B2, v2[0]);
    tp_acc<1, 1, 2>(A1, B1, v2[1]);
    tp_acc<2, 0, 2>(A2, B0, v2[2]);
    tp_acc<2, 2, 2>(A2, B2, v2[3]);
    tp_acc<3, 1, 2>(A3, B1, v2[4]);

    h0[et * 256 + n] = xg[(size_t)e * FIN + n];
    h0[et * 256 + 64 + n * 3 + 0] = sc0[0];
    h0[et * 256 + 64 + n * 3 + 1] = sc1[0];
    h0[et * 256 + 64 + n * 3 + 2] = sc2[0];
#pragma unroll
    for (int p = 0; p < 5; ++p)
#pragma unroll
      for (int i = 0; i < 3; ++i)
        lv1[i * 5120 + et * 320 + p * 64 + n] = v1[p][i];
#pragma unroll
    for (int p = 0; p < 5; ++p)
#pragma unroll
      for (int i = 0; i < 5; ++i)
        lv2[i * 5120 + et * 320 + p * 64 + n] = v2[p][i];
  }
  __syncthreads();

  int wave = tid >> 5, lane = tid & 31, r = lane & 15, h = lane >> 4;

  // ---- phase B1: 8 output GEMMs, one component per wave (K=320, A in LDS) ----
  {
    int c = wave;  // 0..2 -> o1 comps, 3..7 -> o2 comps
    const float* A = (c < 3) ? (lv1 + c * 5120) : (lv2 + (c - 3) * 5120);
    const v2f* pB = (const v2f*)((c < 3) ? pWv1 : pWv2);
    v8f acc[4] = {vzero(), vzero(), vzero(), vzero()};
    for (int k0 = 0; k0 < 320; k0 += 4) {
      v2f a = *(const v2f*)(A + r * 320 + k0 + 2 * h);
      int bbase = (((k0 >> 2) * 2 + h) << 6) + r;
#pragma unroll
      for (int nt = 0; nt < 4; ++nt) {
        v2f b = pB[bbase + nt * 16];
        acc[nt] = wmma4(a, b, acc[nt]);
      }
    }
#pragma unroll
    for (int nt = 0; nt < 4; ++nt)
#pragma unroll
      for (int v = 0; v < 8; ++v) {
        int e = e0 + v + 8 * h;
        int o = nt * 16 + r;
        Vout[(size_t)e * 576 + o * 9 + 1 + c] = acc[nt][v] * 0.05590170f;  // 1/sqrt(320)
        if (c == 0) Vout[(size_t)e * 576 + o * 9] = 0.f;                   // zero l=0 comp
      }
  }
  __syncthreads();

  // ---- phase B2: MLP chain, waves 0..3 each own one N-tile per stage ----
  if (wave < 4) {
    v8f acc[4] = {vzero(), vzero(), vzero(), vzero()};
    const v2f* pB = (const v2f*)pW1;
    for (int k0 = 0; k0 < 256; k0 += 4) {
      v2f a = *(const v2f*)(h0 + r * 256 + k0 + 2 * h);
      int bbase = (((k0 >> 2) * 2 + h) << 6) + r;
#pragma unroll
      for (int nt = 0; nt < 4; ++nt)
        acc[nt] = wmma4(a, pB[bbase + nt * 16], acc[nt]);
    }
    // wave w stores N-tile w (stage output split across waves for parallel store)
    int nt = wave;
#pragma unroll
    for (int v = 0; v < 8; ++v)
      hT[(v + 8 * h) * 64 + nt * 16 + r] = nsilu(acc[nt][v] * 0.0625f);  // /sqrt(256)
  }
  __syncthreads();
  if (wave < 4) {
    v8f acc[4] = {vzero(), vzero(), vzero(), vzero()};
    const v2f* pB = (const v2f*)pW2;
    for (int k0 = 0; k0 < 64; k0 += 4) {
      v2f a = *(const v2f*)(hT + r * 64 + k0 + 2 * h);
      int bbase = (((k0 >> 2) * 2 + h) << 6) + r;
#pragma unroll
      for (int nt = 0; nt < 4; ++nt)
        acc[nt] = wmma4(a, pB[bbase + nt * 16], acc[nt]);
    }
    int nt = wave;
#pragma unroll
    for (int v = 0; v < 8; ++v)
      hT2[(v + 8 * h) * 64 + nt * 16 + r] = nsilu(acc[nt][v] * 0.125f);  // /sqrt(64)
  }
  __syncthreads();
  if (wave < 4) {
    v8f acc[4] = {vzero(), vzero(), vzero(), vzero()};
    const v2f* pB = (const v2f*)pW3;
    for (int k0 = 0; k0 < 64; k0 += 4) {
      v2f a = *(const v2f*)(hT2 + r * 64 + k0 + 2 * h);
      int bbase = (((k0 >> 2) * 2 + h) << 6) + r;
#pragma unroll
      for (int nt = 0; nt < 4; ++nt)
        acc[nt] = wmma4(a, pB[bbase + nt * 16], acc[nt]);
    }
    int nt = wave;
#pragma unroll
    for (int v = 0; v < 8; ++v) {
      int e = e0 + v + 8 * h;
      float d = dws[e];
      float d3 = d * d * d, d6 = d3 * d3, d7 = d6 * d, d8 = d7 * d;
      float u = 1.f - 28.f * d6 + 48.f * d7 - 21.f * d8;
      float fac = (d < 1.f) ? u : 0.f;
      xout[(size_t)e * 64 + nt * 16 + r] = acc[nt][v] * 0.125f * fac;    // /sqrt(64) * env
    }
  }
}

// Note: in phase B2 each of the 4 waves computes all 4 N-tiles but stores only
// its own tile; identical per-wave cost to computing one tile with K-inner
// loops, but keeps A loads at 1 per K-step and all lanes/waves EXEC-full for WMMA.

// ============================== launch ====================================
extern "C" void kernel_launch(void* const* d_in, const int* in_sizes, int n_in,
                              void* d_out, int out_size, void* d_ws, size_t ws_size,
                              hipStream_t stream) {
  const float* vectors = (const float*)d_in[0];
  const float* x       = (const float*)d_in[1];
  const float* V       = (const float*)d_in[2];
  const int*   senders = (const int*)d_in[3];
  const float* W_w     = (const float*)d_in[4];
  const float* W1      = (const float*)d_in[5];
  const float* W2      = (const float*)d_in[6];
  const float* W3      = (const float*)d_in[7];
  const float* Wv1     = (const float*)d_in[8];
  const float* Wv2     = (const float*)d_in[9];

  float* out  = (float*)d_out;
  float* xout = out;                               // (E,64)
  float* Vout = out + (size_t)E_EDGES * 64;        // (E,64,9)

  float* ws      = (float*)d_ws;
  float* nodebuf = ws;                  // 1024*64*16 = 1,048,576 f32
  float* wbuf    = ws + 1048576;        // 16384*64   = 1,048,576 f32
  float* dbuf    = ws + 2097152;        // 16384 f32
  float* pWw     = ws + 2113536;        // 4096
  float* pW1     = ws + 2117632;        // 16384
  float* pW2     = ws + 2134016;        // 4096
  float* pW3     = ws + 2138112;        // 4096
  float* pWv1    = ws + 2142208;        // 20480
  float* pWv2    = ws + 2162688;        // 20480  (end: 2,183,168 floats ~ 8.3 MB)

  k_init<<<dim3((N_NODES * NMUL * 16 + 255) / 256), dim3(256), 0, stream>>>(nodebuf);

  // Pack weights into B-fragment (float2) order
  k_pack<<<dim3((64  / 4 * 128 + 255) / 256), dim3(256), 0, stream>>>(W_w, 64,  pWw);
  k_pack<<<dim3((256 / 4 * 128 + 255) / 256), dim3(256), 0, stream>>>(W1,  256, pW1);
  k_pack<<<dim3((64  / 4 * 128 + 255) / 256), dim3(256), 0, stream>>>(W2,  64,  pW2);
  k_pack<<<dim3((64  / 4 * 128 + 255) / 256), dim3(256), 0, stream>>>(W3,  64,  pW3);
  k_pack<<<dim3((320 / 4 * 128 + 255) / 256), dim3(256), 0, stream>>>(Wv1, 320, pWv1);
  k_pack<<<dim3((320 / 4 * 128 + 255) / 256), dim3(256), 0, stream>>>(Wv2, 320, pWv2);

  k_w<<<dim3(E_EDGES / 16), dim3(32), 0, stream>>>(x, pWw, wbuf);
  k_edge<<<dim3(E_EDGES), dim3(128), 0, stream>>>(vectors, senders, wbuf, nodebuf, dbuf);

  size_t ldsbytes = (size_t)(4096 + 15360 + 25600 + 1024 + 1024) * sizeof(float); // ~184 KB
  k_fused<<<dim3(E_EDGES / 16), dim3(256), ldsbytes, stream>>>(
      x, V, senders, nodebuf, dbuf, pW1, pW2, pW3, pWv1, pWv2, xout, Vout);
}